// Net2_17729624998196
// MI455X (gfx1250) — compile-verified
//
#include <hip/hip_runtime.h>

typedef __attribute__((ext_vector_type(2))) float v2f;
typedef __attribute__((ext_vector_type(8))) float v8f;

#define NNODES 100000
#define NEDGES 1600000

// ---------------------------------------------------------------- zero scratch
__global__ void k_zero(float* __restrict__ p, long n) {
  long i = ((long)blockIdx.x * blockDim.x + threadIdx.x) * 4;
  long stride = (long)gridDim.x * blockDim.x * 4;
  float4 z = {0.f, 0.f, 0.f, 0.f};
  for (long j = i; j < n; j += stride) *(float4*)(p + j) = z;
}

// -------------------------------------------- fold u into edge/node MLP biases
__global__ void k_prep(const float* __restrict__ u,
                       const float* __restrict__ Web, const float* __restrict__ beb,
                       const float* __restrict__ Wnb, const float* __restrict__ bnb,
                       float* __restrict__ bias_eb, float* __restrict__ bias_nb) {
  int j = threadIdx.x;  // 64 threads
  float a = beb[j], b = bnb[j];
  for (int k = 0; k < 64; ++k) {
    float uk = u[k];
    a += uk * Web[(48 + k) * 64 + j];
    b += uk * Wnb[(80 + k) * 64 + j];
  }
  bias_eb[j] = a;
  bias_nb[j] = b;
}

// ------------------------------------------------------------------ edge block
// e_h[16x64] = relu([attr | x[recv] | x[send]] (16x48) @ W_eb[0:48] + bias_eff)
// then fp32 atomic scatter-add into agg_recv / agg_sent and degree counts.
__global__ __launch_bounds__(256) void k_edge(
    const float* __restrict__ x, const float* __restrict__ ea,
    const int* __restrict__ recv, const int* __restrict__ send,
    const float* __restrict__ Web, const float* __restrict__ bias_eb,
    float* __restrict__ agg_r, float* __restrict__ agg_s,
    float* __restrict__ cnt_r, float* __restrict__ cnt_s) {
  __shared__ float sW[12 * 4 * 64];  // 12 k-chunks x 4 n-tiles, fragment layout
  for (int i = threadIdx.x; i < 12 * 4 * 64; i += 256) {
    int j = i & 1, lane = (i >> 1) & 31, nt = (i >> 6) & 3, kc = i >> 8;
    int kk = 4 * kc + 2 * (lane >> 4) + j;
    sW[i] = Web[kk * 64 + nt * 16 + (lane & 15)];
  }
  __syncthreads();
  const int lane = threadIdx.x & 31;
  const int h = lane >> 4, m = lane & 15;
  const int wave = blockIdx.x * 8 + (threadIdx.x >> 5);
  const int nwaves = gridDim.x * 8;
  const int ntiles = NEDGES / 16;
  for (int t = wave; t < ntiles; t += nwaves) {
    const int e0 = t * 16;
    const int er = e0 + m;
    const int ri = recv[er], si = send[er];
    const float* segs[3] = {ea + (long)er * 16, x + (long)ri * 16, x + (long)si * 16};
    v8f acc[4];
#pragma unroll
    for (int nt = 0; nt < 4; ++nt) {
      float bb = bias_eb[nt * 16 + m];
      acc[nt] = (v8f){bb, bb, bb, bb, bb, bb, bb, bb};
    }
#pragma unroll
    for (int seg = 0; seg < 3; ++seg) {
      const float* bp = segs[seg];
#pragma unroll
      for (int cc = 0; cc < 4; ++cc) {
        v2f a = *(const v2f*)(bp + 4 * cc + 2 * h);
        const int kc = seg * 4 + cc;
#pragma unroll
        for (int nt = 0; nt < 4; ++nt) {
          v2f b = *(const v2f*)(&sW[(kc * 4 + nt) * 64 + lane * 2]);
          acc[nt] = __builtin_amdgcn_wmma_f32_16x16x4_f32(false, a, false, b,
                                                          (short)0, acc[nt], false, false);
        }
      }
    }
    int rr[8], ss[8];
#pragma unroll
    for (int v = 0; v < 8; ++v) {
      int e = e0 + 8 * h + v;
      rr[v] = recv[e];
      ss[v] = send[e];
    }
#pragma unroll
    for (int nt = 0; nt < 4; ++nt) {
      const int col = nt * 16 + m;
#pragma unroll
      for (int v = 0; v < 8; ++v) {
        float val = fmaxf(acc[nt][v], 0.0f);  // relu(e_h)
        unsafeAtomicAdd(&agg_r[(long)rr[v] * 64 + col], val);
        unsafeAtomicAdd(&agg_s[(long)ss[v] * 64 + col], val);
      }
    }
    if (lane < 16) {
      unsafeAtomicAdd(&cnt_r[recv[e0 + lane]], 1.0f);
      unsafeAtomicAdd(&cnt_s[send[e0 + lane]], 1.0f);
    }
  }
}

// ------------------------------------------------------------------ node block
// n_h = relu([agg_recv | x] (16x80) @ W_nb[0:80] + bias_eff)
__global__ __launch_bounds__(256) void k_node(
    const float* __restrict__ x, const float* __restrict__ agg_r,
    const float* __restrict__ Wnb, const float* __restrict__ bias_nb,
    float* __restrict__ n_h) {
  __shared__ float sW[20 * 4 * 64];
  for (int i = threadIdx.x; i < 20 * 4 * 64; i += 256) {
    int j = i & 1, lane = (i >> 1) & 31, nt = (i >> 6) & 3, kc = i >> 8;
    int kk = 4 * kc + 2 * (lane >> 4) + j;
    sW[i] = Wnb[kk * 64 + nt * 16 + (lane & 15)];
  }
  __syncthreads();
  const int lane = threadIdx.x & 31;
  const int h = lane >> 4, m = lane & 15;
  const int wave = blockIdx.x * 8 + (threadIdx.x >> 5);
  const int nwaves = gridDim.x * 8;
  const int ntiles = NNODES / 16;
  for (int t = wave; t < ntiles; t += nwaves) {
    const int n0 = t * 16;
    const float* ap = agg_r + (long)(n0 + m) * 64;
    const float* xp = x + (long)(n0 + m) * 16;
    v8f acc[4];
#pragma unroll
    for (int nt = 0; nt < 4; ++nt) {
      float bb = bias_nb[nt * 16 + m];
      acc[nt] = (v8f){bb, bb, bb, bb, bb, bb, bb, bb};
    }
#pragma unroll
    for (int kc = 0; kc < 20; ++kc) {
      const int kk = 4 * kc + 2 * h;
      v2f a = (kc < 16) ? *(const v2f*)(ap + kk) : *(const v2f*)(xp + kk - 64);
#pragma unroll
      for (int nt = 0; nt < 4; ++nt) {
        v2f b = *(const v2f*)(&sW[(kc * 4 + nt) * 64 + lane * 2]);
        acc[nt] = __builtin_amdgcn_wmma_f32_16x16x4_f32(false, a, false, b,
                                                        (short)0, acc[nt], false, false);
      }
    }
#pragma unroll
    for (int nt = 0; nt < 4; ++nt)
#pragma unroll
      for (int v = 0; v < 8; ++v)
        n_h[(long)(n0 + 8 * h + v) * 64 + nt * 16 + m] = fmaxf(acc[nt][v], 0.0f);
  }
}

// ------------------------------------------------- column sum of an [rows,64]
__global__ void k_colsum(const float* __restrict__ src, int rows, float* __restrict__ dst) {
  __shared__ float red[256];
  const int col = threadIdx.x & 63;
  const int g = threadIdx.x >> 6;
  float local = 0.f;
  for (int r = blockIdx.x * 4 + g; r < rows; r += gridDim.x * 4)
    local += src[(long)r * 64 + col];
  red[threadIdx.x] = local;
  __syncthreads();
  if (g == 0) {
    local += red[64 + threadIdx.x] + red[128 + threadIdx.x] + red[192 + threadIdx.x];
    unsafeAtomicAdd(&dst[col], local);
  }
}

// --------------------------------------- global block + fold u2 into d1 bias
__global__ void k_global(const float* __restrict__ u,
                         const float* __restrict__ Wgb, const float* __restrict__ bgb,
                         const float* __restrict__ Wd1, const float* __restrict__ bd1,
                         const float* __restrict__ e_sum, const float* __restrict__ n_sum,
                         float* __restrict__ u2, float* __restrict__ bias_d1) {
  __shared__ float su2[64];
  const int j = threadIdx.x;  // 64 threads
  float acc = bgb[j];
  for (int k = 0; k < 64; ++k) {
    float em = e_sum[k] * (1.0f / NEDGES);
    float nm = n_sum[k] * (1.0f / NNODES);
    acc += em * Wgb[k * 64 + j] + nm * Wgb[(64 + k) * 64 + j] + u[k] * Wgb[(128 + k) * 64 + j];
  }
  acc = fmaxf(acc, 0.0f);
  su2[j] = acc;
  u2[j] = acc;
  __syncthreads();
  float b = bd1[j];
  for (int k = 0; k < 64; ++k) b += su2[k] * Wd1[(192 + k) * 64 + j];
  bias_d1[j] = b;
}

// ---------------------------------------------------------------- decoder
// h = relu([agg_r/cnt | agg_s/cnt | n_h] (16x192) @ W_d1[0:192] + bias_eff)
// out = h @ W_d2 + b_d2   (LDS round-trip to restage h in A-fragment layout)
__global__ __launch_bounds__(256) void k_dec(
    const float* __restrict__ agg_r, const float* __restrict__ agg_s,
    const float* __restrict__ cnt_r, const float* __restrict__ cnt_s,
    const float* __restrict__ n_h, const float* __restrict__ Wd1,
    const float* __restrict__ bias_d1, const float* __restrict__ Wd2,
    const float* __restrict__ bd2, float* __restrict__ out) {
  __shared__ float sW1[48 * 4 * 64];  // 48 KB
  __shared__ float sW2[16 * 64];      // 4 KB
  __shared__ float sH[8 * 16 * 64];   // 32 KB: per-wave hidden tile staging
  for (int i = threadIdx.x; i < 48 * 4 * 64; i += 256) {
    int j = i & 1, lane = (i >> 1) & 31, nt = (i >> 6) & 3, kc = i >> 8;
    int kk = 4 * kc + 2 * (lane >> 4) + j;
    sW1[i] = Wd1[kk * 64 + nt * 16 + (lane & 15)];
  }
  for (int i = threadIdx.x; i < 16 * 64; i += 256) {
    int j = i & 1, lane = (i >> 1) & 31, kc = i >> 6;
    int kk = 4 * kc + 2 * (lane >> 4) + j;
    sW2[i] = Wd2[kk * 16 + (lane & 15)];
  }
  __syncthreads();
  const int lane = threadIdx.x & 31;
  const int h = lane >> 4, m = lane & 15;
  float* hw = sH + (threadIdx.x >> 5) * 1024;  // wave-private 16x64 tile
  const int wave = blockIdx.x * 8 + (threadIdx.x >> 5);
  const int nwaves = gridDim.x * 8;
  const int ntiles = NNODES / 16;
  for (int t = wave; t < ntiles; t += nwaves) {
    const int n0 = t * 16;
    const int n = n0 + m;
    const float invr = 1.0f / fmaxf(cnt_r[n], 1.0f);
    const float invs = 1.0f / fmaxf(cnt_s[n], 1.0f);
    const float* pr = agg_r + (long)n * 64;
    const float* ps = agg_s + (long)n * 64;
    const float* pn = n_h + (long)n * 64;
    v8f acc[4];
#pragma unroll
    for (int nt = 0; nt < 4; ++nt) {
      float bb = bias_d1[nt * 16 + m];
      acc[nt] = (v8f){bb, bb, bb, bb, bb, bb, bb, bb};
    }
#pragma unroll
    for (int kc = 0; kc < 48; ++kc) {
      const int kk = 4 * kc + 2 * h;
      v2f a;
      if (kc < 16)      { a = *(const v2f*)(pr + kk);        a *= invr; }
      else if (kc < 32) { a = *(const v2f*)(ps + kk - 64);   a *= invs; }
      else              { a = *(const v2f*)(pn + kk - 128); }
#pragma unroll
      for (int nt = 0; nt < 4; ++nt) {
        v2f b = *(const v2f*)(&sW1[(kc * 4 + nt) * 64 + lane * 2]);
        acc[nt] = __builtin_amdgcn_wmma_f32_16x16x4_f32(false, a, false, b,
                                                        (short)0, acc[nt], false, false);
      }
    }
    // stage relu(hidden) to LDS (wave-local, no barrier needed)
#pragma unroll
    for (int nt = 0; nt < 4; ++nt)
#pragma unroll
      for (int v = 0; v < 8; ++v)
        hw[(8 * h + v) * 64 + nt * 16 + m] = fmaxf(acc[nt][v], 0.0f);
    // layer 2: 16x64 @ 64x16
    float bb2 = bd2[m];
    v8f acc2 = (v8f){bb2, bb2, bb2, bb2, bb2, bb2, bb2, bb2};
#pragma unroll
    for (int kc = 0; kc < 16; ++kc) {
      const int kk = 4 * kc + 2 * h;
      v2f a = *(const v2f*)(hw + m * 64 + kk);
      v2f b = *(const v2f*)(&sW2[kc * 64 + lane * 2]);
      acc2 = __builtin_amdgcn_wmma_f32_16x16x4_f32(false, a, false, b,
                                                   (short)0, acc2, false, false);
    }
#pragma unroll
    for (int v = 0; v < 8; ++v)
      out[(long)(n0 + 8 * h + v) * 16 + m] = acc2[v];
  }
}

// ---------------------------------------------------------------------- launch
extern "C" void kernel_launch(void* const* d_in, const int* in_sizes, int n_in,
                              void* d_out, int out_size, void* d_ws, size_t ws_size,
                              hipStream_t stream) {
  const float* x    = (const float*)d_in[0];
  const float* ea   = (const float*)d_in[1];
  const float* u    = (const float*)d_in[2];
  const int* send   = (const int*)d_in[3];
  const int* recv   = (const int*)d_in[4];
  const float* Web  = (const float*)d_in[7];
  const float* beb  = (const float*)d_in[8];
  const float* Wnb  = (const float*)d_in[9];
  const float* bnb  = (const float*)d_in[10];
  const float* Wgb  = (const float*)d_in[11];
  const float* bgb  = (const float*)d_in[12];
  const float* Wd1  = (const float*)d_in[13];
  const float* bd1  = (const float*)d_in[14];
  const float* Wd2  = (const float*)d_in[15];
  const float* bd2  = (const float*)d_in[16];
  float* out = (float*)d_out;
  float* ws = (float*)d_ws;

  const long N = NNODES;
  float* agg_r   = ws;                 // N*64
  float* agg_s   = agg_r + N * 64;     // N*64
  float* cnt_r   = agg_s + N * 64;     // N
  float* cnt_s   = cnt_r + N;          // N
  float* e_sum   = cnt_s + N;          // 64
  float* n_sum   = e_sum + 64;         // 64
  float* bias_eb = n_sum + 64;         // 64
  float* bias_nb = bias_eb + 64;       // 64
  float* u2      = bias_nb + 64;       // 64
  float* bias_d1 = u2 + 64;            // 64
  float* n_hbuf  = bias_d1 + 64;       // N*64

  const long zlen = 2 * N * 64 + 2 * N + 128;  // agg + counts + e_sum + n_sum
  k_zero<<<2048, 256, 0, stream>>>(ws, zlen);
  k_prep<<<1, 64, 0, stream>>>(u, Web, beb, Wnb, bnb, bias_eb, bias_nb);
  k_edge<<<2048, 256, 0, stream>>>(x, ea, recv, send, Web, bias_eb,
                                   agg_r, agg_s, cnt_r, cnt_s);
  k_node<<<782, 256, 0, stream>>>(x, agg_r, Wnb, bias_nb, n_hbuf);
  k_colsum<<<512, 256, 0, stream>>>(agg_r, NNODES, e_sum);   // == sum of e_h
  k_colsum<<<512, 256, 0, stream>>>(n_hbuf, NNODES, n_sum);
  k_global<<<1, 64, 0, stream>>>(u, Wgb, bgb, Wd1, bd1, e_sum, n_sum, u2, bias_d1);
  k_dec<<<1024, 256, 0, stream>>>(agg_r, agg_s, cnt_r, cnt_s, n_hbuf,
                                  Wd1, bias_d1, Wd2, bd2, out);
}